// GraphENS_Model_37958920962734
// MI455X (gfx1250) — compile-verified
//
#include <hip/hip_runtime.h>
#include <hip/hip_bf16.h>

typedef __attribute__((ext_vector_type(2))) float v2f;
typedef __attribute__((ext_vector_type(8))) float v8f;

__device__ __forceinline__ v8f wmma_f32(v2f a, v2f b, v8f c) {
    // V_WMMA_F32_16X16X4_F32 : D = A(16x4) x B(4x16) + C(16x16), all f32
    return __builtin_amdgcn_wmma_f32_16x16x4_f32(
        /*neg_a=*/false, a, /*neg_b=*/false, b,
        /*c_mod=*/(short)0, c, /*reuse_a=*/false, /*reuse_b=*/false);
}

// ---------------------------------------------------------------------------
// Layer-1 edge scatter: agg1[dst] += x[src] (128 f32), cnt[dst] += 1.
// One wave per edge; each lane moves 4 contiguous floats (coalesced float4).
// ---------------------------------------------------------------------------
__global__ void scatter_x_kernel(const float* __restrict__ x,
                                 const long long* __restrict__ src,
                                 const long long* __restrict__ dst,
                                 float* __restrict__ agg, float* __restrict__ cnt,
                                 int E) {
    int e = blockIdx.x * (blockDim.x >> 5) + (threadIdx.x >> 5);
    int lane = threadIdx.x & 31;
    if (e >= E) return;
    int s = (int)src[e];
    int d = (int)dst[e];
    const float4 v = ((const float4*)(x + (size_t)s * 128))[lane];
    float* a = agg + (size_t)d * 128 + lane * 4;
    atomicAdd(a + 0, v.x);
    atomicAdd(a + 1, v.y);
    atomicAdd(a + 2, v.z);
    atomicAdd(a + 3, v.w);
    if (lane == 0) atomicAdd(cnt + d, 1.0f);
}

// ---------------------------------------------------------------------------
// Layer-2 edge scatter (64-dim): agg2[dst] += t[src]. One wave per edge,
// 2 floats per lane.
// ---------------------------------------------------------------------------
__global__ void scatter_t_kernel(const float* __restrict__ t,
                                 const long long* __restrict__ src,
                                 const long long* __restrict__ dst,
                                 float* __restrict__ agg2, int E) {
    int e = blockIdx.x * (blockDim.x >> 5) + (threadIdx.x >> 5);
    int lane = threadIdx.x & 31;
    if (e >= E) return;
    int s = (int)src[e];
    int d = (int)dst[e];
    const float2 v = ((const float2*)(t + (size_t)s * 64))[lane];
    float* a = agg2 + (size_t)d * 64 + lane * 2;
    atomicAdd(a + 0, v.x);
    atomicAdd(a + 1, v.y);
}

// ---------------------------------------------------------------------------
// Layer 1: h = relu( (agg1 * rinv) @ W1_l + x @ W1_r + b1 )   [N x 128]
// One wave computes one 16x16 output tile via f32 WMMA, K = 128.
// Mean-scaling (1/max(cnt,1)) folded into the A-operand of the first chain.
// grid = (N/16, 2), block = 128 (4 waves -> tileN = blockIdx.y*4 + wave).
// ---------------------------------------------------------------------------
__global__ void gemm1_kernel(const float* __restrict__ agg,
                             const float* __restrict__ cnt,
                             const float* __restrict__ x,
                             const float* __restrict__ Wl,
                             const float* __restrict__ Wr,
                             const float* __restrict__ b1,
                             float* __restrict__ h) {
    const int wave = threadIdx.x >> 5;
    const int lane = threadIdx.x & 31;
    const int half = lane >> 4;   // 0: K pair {0,1}/rows 0-7 ; 1: K pair {2,3}/rows 8-15
    const int l16  = lane & 15;
    const int m0 = blockIdx.x * 16;
    const int n0 = (blockIdx.y * 4 + wave) * 16;

    const int rowA = m0 + l16;                 // A-operand row for this lane
    const float rinv = 1.0f / fmaxf(cnt[rowA], 1.0f);
    const int col = n0 + l16;                  // B/C column for this lane

    v8f acc;
    const float bias = b1[col];
#pragma unroll
    for (int v = 0; v < 8; ++v) acc[v] = bias;

    const float* aM = agg + (size_t)rowA * 128 + 2 * half;
    const float* aX = x   + (size_t)rowA * 128 + 2 * half;

#pragma unroll 4
    for (int k0 = 0; k0 < 128; k0 += 4) {
        const int kk = k0 + 2 * half;
        v2f a, b;
        // chain 1: mean @ W1_l
        float2 am = *(const float2*)(aM + k0);
        a[0] = am.x * rinv; a[1] = am.y * rinv;
        b[0] = Wl[kk * 128 + col]; b[1] = Wl[(kk + 1) * 128 + col];
        acc = wmma_f32(a, b, acc);
        // chain 2: x @ W1_r
        float2 ax = *(const float2*)(aX + k0);
        a[0] = ax.x; a[1] = ax.y;
        b[0] = Wr[kk * 128 + col]; b[1] = Wr[(kk + 1) * 128 + col];
        acc = wmma_f32(a, b, acc);
    }

#pragma unroll
    for (int v = 0; v < 8; ++v) {
        const int row = m0 + v + 8 * half;
        h[(size_t)row * 128 + col] = fmaxf(acc[v], 0.0f);
    }
}

// ---------------------------------------------------------------------------
// t = h @ W2_l   [N x 64]  (push W2_l through the mean: halves edge traffic)
// grid = (N/16, 1), block = 128 (4 waves -> 4 N-tiles of 16).
// ---------------------------------------------------------------------------
__global__ void gemm_t_kernel(const float* __restrict__ h,
                              const float* __restrict__ Wl,
                              float* __restrict__ t) {
    const int wave = threadIdx.x >> 5;
    const int lane = threadIdx.x & 31;
    const int half = lane >> 4;
    const int l16  = lane & 15;
    const int m0 = blockIdx.x * 16;
    const int n0 = wave * 16;

    const int rowA = m0 + l16;
    const int col = n0 + l16;

    v8f acc = {};
    const float* aH = h + (size_t)rowA * 128 + 2 * half;

#pragma unroll 4
    for (int k0 = 0; k0 < 128; k0 += 4) {
        const int kk = k0 + 2 * half;
        v2f a, b;
        float2 ah = *(const float2*)(aH + k0);
        a[0] = ah.x; a[1] = ah.y;
        b[0] = Wl[kk * 64 + col]; b[1] = Wl[(kk + 1) * 64 + col];
        acc = wmma_f32(a, b, acc);
    }

#pragma unroll
    for (int v = 0; v < 8; ++v) {
        const int row = m0 + v + 8 * half;
        t[(size_t)row * 64 + col] = acc[v];
    }
}

// ---------------------------------------------------------------------------
// out = agg2 * rinv + h @ W2_r + b2   [N x 64]
// Mean term + bias loaded directly into the WMMA C accumulator.
// grid = (N/16, 1), block = 128.
// ---------------------------------------------------------------------------
__global__ void gemm_out_kernel(const float* __restrict__ agg2,
                                const float* __restrict__ cnt,
                                const float* __restrict__ h,
                                const float* __restrict__ Wr,
                                const float* __restrict__ b2,
                                float* __restrict__ out) {
    const int wave = threadIdx.x >> 5;
    const int lane = threadIdx.x & 31;
    const int half = lane >> 4;
    const int l16  = lane & 15;
    const int m0 = blockIdx.x * 16;
    const int n0 = wave * 16;

    const int rowA = m0 + l16;
    const int col = n0 + l16;

    v8f acc;
    const float bias = b2[col];
#pragma unroll
    for (int v = 0; v < 8; ++v) {
        const int row = m0 + v + 8 * half;
        const float rv = 1.0f / fmaxf(cnt[row], 1.0f);
        acc[v] = agg2[(size_t)row * 64 + col] * rv + bias;
    }

    const float* aH = h + (size_t)rowA * 128 + 2 * half;
#pragma unroll 4
    for (int k0 = 0; k0 < 128; k0 += 4) {
        const int kk = k0 + 2 * half;
        v2f a, b;
        float2 ah = *(const float2*)(aH + k0);
        a[0] = ah.x; a[1] = ah.y;
        b[0] = Wr[kk * 64 + col]; b[1] = Wr[(kk + 1) * 64 + col];
        acc = wmma_f32(a, b, acc);
    }

#pragma unroll
    for (int v = 0; v < 8; ++v) {
        const int row = m0 + v + 8 * half;
        out[(size_t)row * 64 + col] = acc[v];
    }
}

// ---------------------------------------------------------------------------
extern "C" void kernel_launch(void* const* d_in, const int* in_sizes, int n_in,
                              void* d_out, int out_size, void* d_ws, size_t ws_size,
                              hipStream_t stream) {
    const float*     x    = (const float*)d_in[0];
    const long long* ei   = (const long long*)d_in[1];   // int64 [2, E]
    const float*     W1_l = (const float*)d_in[2];       // [128,128]
    const float*     W1_r = (const float*)d_in[3];       // [128,128]
    const float*     b1   = (const float*)d_in[4];       // [128]
    const float*     W2_l = (const float*)d_in[5];       // [128,64]
    const float*     W2_r = (const float*)d_in[6];       // [128,64]
    const float*     b2   = (const float*)d_in[7];       // [64]
    float*           out  = (float*)d_out;               // [N,64]

    const int N = in_sizes[0] / 128;
    const int E = in_sizes[1] / 2;
    const long long* src = ei;
    const long long* dst = ei + E;

    // Workspace layout (floats): agg1[N*128] | h[N*128] | t2[N*64] | agg2[N*64] | cnt[N]
    float* ws   = (float*)d_ws;
    float* agg1 = ws;
    float* h    = agg1 + (size_t)N * 128;
    float* t2   = h    + (size_t)N * 128;
    float* agg2 = t2   + (size_t)N * 64;
    float* cnt  = agg2 + (size_t)N * 64;

    hipMemsetAsync(agg1, 0, (size_t)N * 128 * sizeof(float), stream);
    hipMemsetAsync(agg2, 0, (size_t)N * 64  * sizeof(float), stream);
    hipMemsetAsync(cnt,  0, (size_t)N       * sizeof(float), stream);

    // Layer 1: neighbor sum + degree
    {
        const int wavesPerBlock = 8;  // 256 threads
        const int blocks = (E + wavesPerBlock - 1) / wavesPerBlock;
        scatter_x_kernel<<<blocks, 256, 0, stream>>>(x, src, dst, agg1, cnt, E);
    }
    // h = relu(mean @ W1_l + x @ W1_r + b1)
    gemm1_kernel<<<dim3(N / 16, 2), 128, 0, stream>>>(agg1, cnt, x, W1_l, W1_r, b1, h);
    // t = h @ W2_l  (aggregate in 64-dim instead of 128-dim)
    gemm_t_kernel<<<dim3(N / 16, 1), 128, 0, stream>>>(h, W2_l, t2);
    // Layer 2: neighbor sum of t
    {
        const int blocks = (E + 7) / 8;
        scatter_t_kernel<<<blocks, 256, 0, stream>>>(t2, src, dst, agg2, E);
    }
    // out = mean(t) + h @ W2_r + b2
    gemm_out_kernel<<<dim3(N / 16, 1), 128, 0, stream>>>(agg2, cnt, h, W2_r, b2, out);
}